// EncoderDecoder_75668733821148
// MI455X (gfx1250) — compile-verified
//
#include <hip/hip_runtime.h>
#include <hip/hip_bf16.h>

// ---------------------------------------------------------------------------
// Model dims (fixed by the reference)
// ---------------------------------------------------------------------------
#define NN   16384      // nodes
#define EE   262144     // edges
#define BB   64         // batch
#define TT   12         // time steps
#define INF  64         // input features
#define REGF 200
#define REF  1600       // region edges

typedef float v2f __attribute__((ext_vector_type(2)));
typedef float v8f __attribute__((ext_vector_type(8)));
typedef unsigned int u32x4 __attribute__((ext_vector_type(4)));
typedef int i32x4 __attribute__((ext_vector_type(4)));
typedef int i32x8 __attribute__((ext_vector_type(8)));

#if defined(__AMDGCN__) && __has_builtin(__builtin_amdgcn_tensor_load_to_lds) && \
    __has_builtin(__builtin_amdgcn_s_wait_tensorcnt)
#define HAVE_TDM 1
#endif

__device__ __forceinline__ float gelu_f(float x) {
    return 0.5f * x * (1.0f + erff(x * 0.70710678118654752f));
}
__device__ __forceinline__ float sigm_f(float x) {
    return 1.0f / (1.0f + expf(-x));
}

// ---------------------------------------------------------------------------
// Elementwise / utility kernels
// ---------------------------------------------------------------------------
__global__ void k_zero(float* __restrict__ p, size_t n) {
    size_t i = (size_t)blockIdx.x * blockDim.x + threadIdx.x;
    if (i < n) p[i] = 0.0f;
}

__global__ void k_edge_degree(const int* __restrict__ src, const int* __restrict__ dst,
                              float* __restrict__ deg, int E, int nnodes) {
    int i = blockIdx.x * blockDim.x + threadIdx.x;
    if (i >= E) return;
    atomicAdd(&deg[src[i]], 1.0f);
    atomicAdd(&deg[nnodes + dst[i]], 1.0f);
}

__global__ void k_rsqrt_clamp(float* __restrict__ d, int n) {
    int i = blockIdx.x * blockDim.x + threadIdx.x;
    if (i < n) d[i] = rsqrtf(fmaxf(d[i], 1.0f));
}

__global__ void k_counts(const int* __restrict__ lbi, float* __restrict__ cnt, int n) {
    int i = blockIdx.x * blockDim.x + threadIdx.x;
    if (i < n) atomicAdd(&cnt[lbi[i]], 1.0f);
}

__global__ void k_inv_clamp(float* __restrict__ d, int n) {
    int i = blockIdx.x * blockDim.x + threadIdx.x;
    if (i < n) d[i] = 1.0f / fmaxf(d[i], 1.0f);
}

__global__ void k_transpose(const float* __restrict__ W, float* __restrict__ WT, int R, int C) {
    int i = blockIdx.x * blockDim.x + threadIdx.x;
    if (i >= R * C) return;
    int r = i / C, c = i % C;
    WT[(size_t)c * R + r] = W[(size_t)r * C + c];
}

__global__ void k_add_vec(const float* __restrict__ a, const float* __restrict__ b,
                          float* __restrict__ o, int n) {
    int i = blockIdx.x * blockDim.x + threadIdx.x;
    if (i < n) o[i] = a[i] + b[i];
}

__global__ void k_scale(const float* __restrict__ in, float* __restrict__ out, float s, size_t n) {
    size_t i = (size_t)blockIdx.x * blockDim.x + threadIdx.x;
    if (i < n) out[i] = in[i] * s;
}

// ---------------------------------------------------------------------------
// BatchNorm: stats (sum, sumsq per column) then apply
// ---------------------------------------------------------------------------
__global__ void k_bn_stats(const float* __restrict__ x, float* __restrict__ stats,
                           int R, int C, int rowsPerBlock) {
    extern __shared__ float sm[];          // 2*C floats
    float* ssum = sm;
    float* ssq  = sm + C;
    for (int i = threadIdx.x; i < 2 * C; i += blockDim.x) sm[i] = 0.0f;
    __syncthreads();
    int r0 = blockIdx.x * rowsPerBlock;
    int r1 = min(r0 + rowsPerBlock, R);
    int cnt = (r1 - r0) * C;
    const float* base = x + (size_t)r0 * C;
    for (int i = threadIdx.x; i < cnt; i += blockDim.x) {
        int f = i % C;
        float v = base[i];
        atomicAdd(&ssum[f], v);
        atomicAdd(&ssq[f], v * v);
    }
    __syncthreads();
    for (int f = threadIdx.x; f < C; f += blockDim.x) {
        atomicAdd(&stats[f], ssum[f]);
        atomicAdd(&stats[C + f], ssq[f]);
    }
}

__global__ void k_bn_apply(const float* __restrict__ x, float* __restrict__ y,
                           const float* __restrict__ stats,
                           const float* __restrict__ gamma, const float* __restrict__ beta,
                           int R, int C) {
    size_t i = (size_t)blockIdx.x * blockDim.x + threadIdx.x;
    if (i >= (size_t)R * C) return;
    int f = (int)(i % C);
    float mean = stats[f] / (float)R;
    float var  = fmaxf(stats[C + f] / (float)R - mean * mean, 0.0f);
    y[i] = (x[i] - mean) * rsqrtf(var + 1e-5f) * gamma[f] + beta[f];
}

// ---------------------------------------------------------------------------
// Edge scatter with both GCN degree scales folded in:
//   m[dst] += x[src] * s_src[src] * s_dst[dst]
// (s_dst is constant per output row of the segment-sum, so folding is exact)
// ---------------------------------------------------------------------------
__global__ void k_scatter(const float* __restrict__ x, int lda,
                          const float* __restrict__ s_src,
                          const float* __restrict__ s_dst,
                          const int* __restrict__ src, const int* __restrict__ dst,
                          float* __restrict__ m, int E, int C) {
    size_t i = (size_t)blockIdx.x * blockDim.x + threadIdx.x;
    size_t total = (size_t)E * (size_t)C;
    if (i >= total) return;
    int e = (int)(i / C);
    int f = (int)(i % C);
    int s = src[e];
    int d = dst[e];
    float v = x[(size_t)s * lda + f] * s_src[s] * s_dst[d];
    atomicAdd(&m[(size_t)d * C + f], v);
}

// ---------------------------------------------------------------------------
// WMMA fp32 GEMM: C[M,N] = act( A[M,K] * B[K,N] + bias + addend )
// Block = 256 threads = 8 waves (4 M-groups x 2 N-groups).
// Block tile 128x64, wave tile 32x32 (4 v8f accumulators), K chunk 32.
// Rows >= M are zero-padded/guarded; N must be a multiple of 64.
// ---------------------------------------------------------------------------
__global__ __launch_bounds__(256)
void k_gemm_wmma(const float* __restrict__ A, int lda,
                 const float* __restrict__ B,
                 const float* __restrict__ bias,
                 const float* __restrict__ addend,
                 float* __restrict__ C,
                 int M, int K, int N, int act) {
    __shared__ float As[128][34];
    __shared__ float Bs[32][66];
    const int tid  = threadIdx.x;
    const int lane = tid & 31;
    const int wave = tid >> 5;
    const int wm = wave >> 1;              // 0..3 (M group of 32)
    const int wn = wave & 1;               // 0..1 (N group of 32)
    const int ml = lane & 15;
    const int kb = (lane >> 4) << 1;       // 0 or 2 (K pair select)
    const int m0 = blockIdx.y * 128;
    const int n0 = blockIdx.x * 64;

    v8f acc00 = {}, acc01 = {}, acc10 = {}, acc11 = {};
    for (int kc = 0; kc < K; kc += 32) {
        // stage A tile 128x32 (16 elems / thread), zero-pad M and K tails
#pragma unroll
        for (int i = 0; i < 16; ++i) {
            int idx = tid + i * 256;
            int r = idx >> 5;
            int c = idx & 31;
            int gm = m0 + r;
            int gk = kc + c;
            As[r][c] = (gm < M && gk < K) ? A[(size_t)gm * lda + gk] : 0.0f;
        }
        // stage B tile 32x64 (8 elems / thread), zero-pad K tail
#pragma unroll
        for (int i = 0; i < 8; ++i) {
            int idx = tid + i * 256;
            int r = idx >> 6;
            int c = idx & 63;
            int gk = kc + r;
            Bs[r][c] = (gk < K) ? B[(size_t)gk * N + n0 + c] : 0.0f;
        }
        // prefetch next A chunk (gfx1250 global_prefetch path)
        if (kc + 32 < K) {
            int pm = m0 + (tid >> 1);
            if (pm < M) __builtin_prefetch(&A[(size_t)pm * lda + kc + 32], 0, 1);
        }
        __syncthreads();
        const int ar0 = wm * 32 + ml;
        const int bc0 = wn * 32 + ml;
#pragma unroll
        for (int kk = 0; kk < 32; kk += 4) {
            v2f a0, a1, b0, b1;
            a0[0] = As[ar0][kk + kb];
            a0[1] = As[ar0][kk + kb + 1];
            a1[0] = As[ar0 + 16][kk + kb];
            a1[1] = As[ar0 + 16][kk + kb + 1];
            b0[0] = Bs[kk + kb][bc0];
            b0[1] = Bs[kk + kb + 1][bc0];
            b1[0] = Bs[kk + kb][bc0 + 16];
            b1[1] = Bs[kk + kb + 1][bc0 + 16];
            acc00 = __builtin_amdgcn_wmma_f32_16x16x4_f32(false, a0, false, b0, (short)0, acc00, false, false);
            acc01 = __builtin_amdgcn_wmma_f32_16x16x4_f32(false, a0, false, b1, (short)0, acc01, false, false);
            acc10 = __builtin_amdgcn_wmma_f32_16x16x4_f32(false, a1, false, b0, (short)0, acc10, false, false);
            acc11 = __builtin_amdgcn_wmma_f32_16x16x4_f32(false, a1, false, b1, (short)0, acc11, false, false);
        }
        __syncthreads();
    }
    // epilogue
    const int nA = n0 + wn * 32 + ml;
    const int nB = nA + 16;
    const float bvA = bias ? bias[nA] : 0.0f;
    const float bvB = bias ? bias[nB] : 0.0f;
    const int mbase = m0 + wm * 32 + ((lane >> 4) << 3);
#pragma unroll
    for (int i = 0; i < 8; ++i) {
        int r0 = mbase + i;
        int r1 = r0 + 16;
        if (r0 < M) {
            float v0 = acc00[i] + bvA;
            float v1 = acc01[i] + bvB;
            if (addend) {
                v0 += addend[(size_t)r0 * N + nA];
                v1 += addend[(size_t)r0 * N + nB];
            }
            if (act == 1) { v0 = gelu_f(v0); v1 = gelu_f(v1); }
            C[(size_t)r0 * N + nA] = v0;
            C[(size_t)r0 * N + nB] = v1;
        }
        if (r1 < M) {
            float v0 = acc10[i] + bvA;
            float v1 = acc11[i] + bvB;
            if (addend) {
                v0 += addend[(size_t)r1 * N + nA];
                v1 += addend[(size_t)r1 * N + nB];
            }
            if (act == 1) { v0 = gelu_f(v0); v1 = gelu_f(v1); }
            C[(size_t)r1 * N + nA] = v0;
            C[(size_t)r1 * N + nB] = v1;
        }
    }
}

// ---------------------------------------------------------------------------
// TDM warm-up: DMA a 32x64 tile of W into LDS via the Tensor Data Mover.
// Pure prefetch (L2 warm + exercises tensor_load_to_lds / s_wait_tensorcnt);
// no global side effects.
// ---------------------------------------------------------------------------
__global__ void k_tdm_warm(const float* __restrict__ W, int rows, int cols) {
#ifdef HAVE_TDM
    __shared__ float tile[32 * 64];
    unsigned long long ga = (unsigned long long)(const void*)W;
    unsigned int ldsOff =
        (unsigned int)(unsigned long long)(__attribute__((address_space(3))) float*)&tile[0];
    u32x4 g0;
    g0[0] = 1u;                                            // count=1
    g0[1] = ldsOff;                                        // lds_addr (bytes)
    g0[2] = (unsigned int)(ga & 0xffffffffull);            // global_addr[31:0]
    g0[3] = (unsigned int)((ga >> 32) & 0x1ffffffull) | (2u << 30);  // [56:32] | type=2
    i32x8 g1;
    unsigned ucols = (unsigned)cols, urows = (unsigned)rows;
    g1[0] = (int)(2u << 16);                               // data_size=4B
    g1[1] = (int)((ucols & 0xffffu) << 16);                // tensor_dim0[15:0]
    g1[2] = (int)(((ucols >> 16) & 0xffffu) | ((urows & 0xffffu) << 16));  // dim0 hi | dim1 lo
    g1[3] = (int)(((urows >> 16) & 0xffffu) | (64u << 16));// dim1 hi | tile_dim0=64
    g1[4] = 32;                                            // tile_dim1=32, tile_dim2=0
    g1[5] = (int)ucols;                                    // tensor_dim0_stride[31:0]
    g1[6] = 0;
    g1[7] = 0;
    i32x4 z4 = {};
    if ((threadIdx.x >> 5) == 0) {
#if __clang_major__ >= 23
        i32x8 z8 = {};
        __builtin_amdgcn_tensor_load_to_lds(g0, g1, z4, z4, z8, 0);
#else
        __builtin_amdgcn_tensor_load_to_lds(g0, g1, z4, z4, 0);
#endif
        __builtin_amdgcn_s_wait_tensorcnt(0);
    }
    __syncthreads();
    if (tile[threadIdx.x] == 1234567.0f) __threadfence();  // keep tile live
#else
    (void)W; (void)rows; (void)cols;
#endif
}

// ---------------------------------------------------------------------------
// Small fallback GEMM (odd shapes): thread per output element
// ---------------------------------------------------------------------------
__global__ void k_small_gemm(const float* __restrict__ A, int lda,
                             const float* __restrict__ B,
                             const float* __restrict__ bias,
                             float* __restrict__ C,
                             int M, int K, int N, int act) {
    int i = blockIdx.x * blockDim.x + threadIdx.x;
    if (i >= M * N) return;
    int r = i / N, c = i % N;
    float acc = bias ? bias[c] : 0.0f;
    const float* ar = A + (size_t)r * lda;
    for (int k = 0; k < K; ++k) acc += ar[k] * B[(size_t)k * N + c];
    if (act == 1) acc = gelu_f(acc);
    C[i] = acc;
}

// Fused small LSTM gate kernel: gates = x@WihT + h@WhhT + bias
__global__ void k_small_gates(const float* __restrict__ x, int lda, int Kx,
                              const float* __restrict__ WihT,
                              const float* __restrict__ h, int H,
                              const float* __restrict__ WhhT,
                              const float* __restrict__ bias,
                              float* __restrict__ gates, int M, int G) {
    int i = blockIdx.x * blockDim.x + threadIdx.x;
    if (i >= M * G) return;
    int r = i / G, j = i % G;
    float acc = bias[j];
    const float* xr = x + (size_t)r * lda;
    for (int k = 0; k < Kx; ++k) acc += xr[k] * WihT[(size_t)k * G + j];
    const float* hr = h + (size_t)r * H;
    for (int k = 0; k < H; ++k) acc += hr[k] * WhhT[(size_t)k * G + j];
    gates[i] = acc;
}

// LSTM cell: gates (M x 4H) -> update c, h; optional seq write and accumulator
__global__ void k_lstm_cell(const float* __restrict__ gates,
                            float* __restrict__ c, float* __restrict__ h,
                            float* __restrict__ seq, int seq_ld,
                            float* __restrict__ acc,
                            int M, int H) {
    int i = blockIdx.x * blockDim.x + threadIdx.x;
    if (i >= M * H) return;
    int r = i / H, j = i % H;
    const float* g = gates + (size_t)r * 4 * H;
    float ig = sigm_f(g[j]);
    float fg = sigm_f(g[H + j]);
    float gg = tanhf(g[2 * H + j]);
    float og = sigm_f(g[3 * H + j]);
    float cc = fg * c[i] + ig * gg;
    float hh = og * tanhf(cc);
    c[i] = cc;
    h[i] = hh;
    if (seq) seq[(size_t)r * seq_ld + j] = hh;
    if (acc) acc[i] += hh;
}

// ---------------------------------------------------------------------------
// Assembly / gather kernels
// ---------------------------------------------------------------------------
__global__ void k_build_ctx(const int* __restrict__ weeks, const int* __restrict__ minutes,
                            const float* __restrict__ wemb, const float* __restrict__ memb,
                            float* __restrict__ ctx, int BT) {
    int i = blockIdx.x * blockDim.x + threadIdx.x;
    if (i >= BT * 66) return;
    int bt = i / 66, f = i % 66;
    ctx[i] = (f < 6) ? wemb[weeks[bt] * 6 + f] : memb[minutes[bt] * 60 + (f - 6)];
}

__global__ void k_mean_T(const float* __restrict__ seq, float* __restrict__ out,
                         int Bn, int T, int H) {
    int i = blockIdx.x * blockDim.x + threadIdx.x;
    if (i >= Bn * H) return;
    int b = i / H, j = i % H;
    float s = 0.0f;
    for (int t = 0; t < T; ++t) s += seq[((size_t)b * T + t) * H + j];
    out[i] = s / (float)T;
}

__global__ void k_gather_rows(const float* __restrict__ table, const int* __restrict__ idx,
                              float* __restrict__ out, int n, int C) {
    size_t i = (size_t)blockIdx.x * blockDim.x + threadIdx.x;
    if (i >= (size_t)n * C) return;
    int r = (int)(i / C), f = (int)(i % C);
    out[i] = table[(size_t)idx[r] * C + f];
}

__global__ void k_build_xl(const float* __restrict__ lsf, const float* __restrict__ lemb,
                           const int* __restrict__ lsi, float* __restrict__ xl, int n) {
    size_t i = (size_t)blockIdx.x * blockDim.x + threadIdx.x;
    if (i >= (size_t)n * 96) return;
    int r = (int)(i / 96), f = (int)(i % 96);
    xl[i] = (f < 32) ? lsf[(size_t)r * 32 + f]
                     : lemb[(size_t)lsi[r] * 64 + (f - 32)];
}

__global__ void k_build_rnn_in(const float* __restrict__ hs, const float* __restrict__ lt,
                               const float* __restrict__ gsp, float* __restrict__ out,
                               int Bn, int T) {
    int i = blockIdx.x * blockDim.x + threadIdx.x;
    if (i >= Bn * T * 252) return;
    int bt = i / 252, f = i % 252;
    int b = bt / T;
    float v;
    if (f < 128)      v = hs[(size_t)bt * 128 + f];
    else if (f < 188) v = lt[(size_t)bt * 60 + (f - 128)];
    else              v = gsp[(size_t)b * 64 + (f - 188)];
    out[i] = v;
}

__global__ void k_build_y(const float* __restrict__ ht, const float* __restrict__ gth,
                          const float* __restrict__ lsh, float* __restrict__ y, int n) {
    size_t i = (size_t)blockIdx.x * blockDim.x + threadIdx.x;
    if (i >= (size_t)n * 252) return;
    int r = (int)(i / 252), f = (int)(i % 252);
    float v;
    if (f < 128)      v = ht[(size_t)r * 128 + f];
    else if (f < 188) v = gth[(size_t)r * 60 + (f - 128)];
    else              v = lsh[(size_t)r * 64 + (f - 188)];
    y[i] = v;
}

__global__ void k_seg_accum(const float* __restrict__ x, const int* __restrict__ lbi,
                            const float* __restrict__ invc, float* __restrict__ hs,
                            int n, int T, int t, int C) {
    size_t i = (size_t)blockIdx.x * blockDim.x + threadIdx.x;
    if (i >= (size_t)n * C) return;
    int r = (int)(i / C), f = (int)(i % C);
    int b = lbi[r];
    atomicAdd(&hs[((size_t)b * T + t) * C + f], x[i] * invc[b]);
}

__global__ void k_build_ab(const float* __restrict__ aseq, const float* __restrict__ hs,
                           const float* __restrict__ yb, const float* __restrict__ ht,
                           const int* __restrict__ lbi, float* __restrict__ ab,
                           int n, int T) {
    size_t i = (size_t)blockIdx.x * blockDim.x + threadIdx.x;
    if (i >= (size_t)n * T * 256) return;
    int nn = (int)(i / (T * 256));
    int rem = (int)(i % (T * 256));
    int t = rem / 256, f = rem % 256;
    float v;
    if (f < 128) {
        int b = lbi[nn];
        size_t o = ((size_t)b * T + t) * 128 + f;
        v = aseq[o] + hs[o];
    } else {
        size_t o = (size_t)nn * 128 + (f - 128);
        v = yb[o] + ht[o];
    }
    ab[i] = v;
}

// ---------------------------------------------------------------------------
// Host-side helpers
// ---------------------------------------------------------------------------
static inline int cdiv(long long a, long long b) { return (int)((a + b - 1) / b); }

static float* wsAlloc(char*& p, size_t nfloats) {
    float* r = (float*)p;
    p += ((nfloats * sizeof(float) + 255) / 256) * 256;
    return r;
}

static void zero(hipStream_t s, float* p, size_t n) {
    k_zero<<<cdiv((long long)n, 256), 256, 0, s>>>(p, n);
}

static void gemm(hipStream_t s, const float* A, int lda, const float* B,
                 const float* bias, const float* addend,
                 float* C, int M, int K, int N, int act) {
    dim3 g(N / 64, cdiv(M, 128));
    k_gemm_wmma<<<g, 256, 0, s>>>(A, lda, B, bias, addend, C, M, K, N, act);
}

static void bn(hipStream_t s, float* stats, const float* x, float* y,
               const float* gamma, const float* beta, int R, int C) {
    zero(s, stats, 2 * C);
    int rpb = 64;
    k_bn_stats<<<cdiv(R, rpb), 256, 2 * C * sizeof(float), s>>>(x, stats, R, C, rpb);
    size_t total = (size_t)R * C;
    k_bn_apply<<<cdiv((long long)total, 256), 256, 0, s>>>(x, y, stats, gamma, beta, R, C);
}

static void scatter(hipStream_t s, const float* x, int lda,
                    const float* s_src, const float* s_dst,
                    const int* src, const int* dst, float* m, int E, int C) {
    size_t total = (size_t)E * C;
    k_scatter<<<cdiv((long long)total, 256), 256, 0, s>>>(x, lda, s_src, s_dst, src, dst, m, E, C);
}

// ---------------------------------------------------------------------------
// kernel_launch
// ---------------------------------------------------------------------------
extern "C" void kernel_launch(void* const* d_in, const int* in_sizes, int n_in,
                              void* d_out, int out_size, void* d_ws, size_t ws_size,
                              hipStream_t stream) {
    (void)in_sizes; (void)n_in; (void)out_size; (void)ws_size;

    // ---- inputs ----
    const int*   weeks   = (const int*)d_in[0];
    const int*   minutes = (const int*)d_in[1];
    const int*   gsi     = (const int*)d_in[2];
    const int*   src     = (const int*)d_in[3];
    const int*   dst     = (const int*)d_in[4];
    const float* traffic = (const float*)d_in[5];
    const int*   lbi     = (const int*)d_in[6];
    const int*   lsi     = (const int*)d_in[7];
    const float* lsf     = (const float*)d_in[8];
    const int*   gsrc    = (const int*)d_in[9];
    const int*   gdst    = (const int*)d_in[10];
    // params: sorted-key pytree leaves starting at d_in[11]
    auto P = [&](int i) { return (const float*)d_in[11 + i]; };
    // 0 bn1.g 1 bn1.b | 2..9 ctx_lstm | 10,11 dec_bn1 | 12,13 dec_bn2
    // 14..17 dec_conv | 18..25 dec_lstm | 26,27 enc_bn | 28..31 enc_conv
    // 32..39 enc_lstm | 40..43 fusion | 44,45 global_gcn | 46 link_emb
    // 47,48 local_gcn | 49 minute_emb | 50 region_emb | 51 week_emb

    float* out = (float*)d_out;

    // ---- workspace ----
    char* wp = (char*)d_ws;
    float* deg    = wsAlloc(wp, 2 * NN);
    float* gdeg   = wsAlloc(wp, 2 * REGF);
    float* cnts   = wsAlloc(wp, BB);
    float* stats  = wsAlloc(wp, 512);
    float* h_bn   = wsAlloc(wp, (size_t)NN * TT * INF);
    float* mbuf   = wsAlloc(wp, (size_t)NN * 256);
    float* xa     = wsAlloc(wp, (size_t)NN * 128);
    float* xb     = wsAlloc(wp, (size_t)NN * 128);
    float* hs     = wsAlloc(wp, (size_t)BB * TT * 128);
    float* gates  = wsAlloc(wp, (size_t)NN * 512);
    float* h1     = wsAlloc(wp, (size_t)NN * 128);
    float* c1     = wsAlloc(wp, (size_t)NN * 128);
    float* h2     = wsAlloc(wp, (size_t)NN * 128);
    float* c2     = wsAlloc(wp, (size_t)NN * 128);
    float* htacc  = wsAlloc(wp, (size_t)NN * 128);
    float* ht     = wsAlloc(wp, (size_t)NN * 128);
    float* ctxin  = wsAlloc(wp, (size_t)BB * TT * 66);
    float* lt     = wsAlloc(wp, (size_t)BB * TT * 60);
    float* cg     = wsAlloc(wp, (size_t)BB * 240);
    float* ch1    = wsAlloc(wp, BB * 60);
    float* cc1    = wsAlloc(wp, BB * 60);
    float* ch2    = wsAlloc(wp, BB * 60);
    float* cc2    = wsAlloc(wp, BB * 60);
    float* gtm    = wsAlloc(wp, BB * 60);
    float* gm     = wsAlloc(wp, REGF * 64);
    float* gemb   = wsAlloc(wp, REGF * 64);
    float* gsp    = wsAlloc(wp, BB * 64);
    float* gth    = wsAlloc(wp, (size_t)NN * 60);
    float* xl     = wsAlloc(wp, (size_t)NN * 96);
    float* lsh    = wsAlloc(wp, (size_t)NN * 64);
    float* rnnin  = wsAlloc(wp, (size_t)BB * TT * 252);
    float* ybuf   = wsAlloc(wp, (size_t)NN * 252);
    float* aseq   = wsAlloc(wp, (size_t)BB * TT * 128);
    float* dg     = wsAlloc(wp, (size_t)BB * 512);
    float* dh1    = wsAlloc(wp, BB * 128);
    float* dc1    = wsAlloc(wp, BB * 128);
    float* dh2    = wsAlloc(wp, BB * 128);
    float* dc2    = wsAlloc(wp, BB * 128);
    float* fust   = wsAlloc(wp, (size_t)24576 * 256);
    float* eWihT1 = wsAlloc(wp, 64 * 512);
    float* eWhhT1 = wsAlloc(wp, 128 * 512);
    float* eWihT2 = wsAlloc(wp, 128 * 512);
    float* eWhhT2 = wsAlloc(wp, 128 * 512);
    float* dWihT1 = wsAlloc(wp, 252 * 512);
    float* dWhhT1 = wsAlloc(wp, 128 * 512);
    float* dWihT2 = wsAlloc(wp, 128 * 512);
    float* dWhhT2 = wsAlloc(wp, 128 * 512);
    float* cWihT1 = wsAlloc(wp, 66 * 240);
    float* cWhhT1 = wsAlloc(wp, 60 * 240);
    float* cWihT2 = wsAlloc(wp, 60 * 240);
    float* cWhhT2 = wsAlloc(wp, 60 * 240);
    float* bs_e0  = wsAlloc(wp, 512);
    float* bs_e1  = wsAlloc(wp, 512);
    float* bs_d0  = wsAlloc(wp, 512);
    float* bs_d1  = wsAlloc(wp, 512);
    float* bs_c0  = wsAlloc(wp, 240);
    float* bs_c1  = wsAlloc(wp, 240);

    // ---- phase 0: degrees, counts, weight prep ----
    zero(stream, deg, 2 * NN);
    k_edge_degree<<<cdiv(EE, 256), 256, 0, stream>>>(src, dst, deg, EE, NN);
    k_rsqrt_clamp<<<cdiv(2 * NN, 256), 256, 0, stream>>>(deg, 2 * NN);
    zero(stream, gdeg, 2 * REGF);
    k_edge_degree<<<cdiv(REF, 256), 256, 0, stream>>>(gsrc, gdst, gdeg, REF, REGF);
    k_rsqrt_clamp<<<cdiv(2 * REGF, 256), 256, 0, stream>>>(gdeg, 2 * REGF);
    zero(stream, cnts, BB);
    k_counts<<<cdiv(NN, 256), 256, 0, stream>>>(lbi, cnts, NN);
    k_inv_clamp<<<1, 256, 0, stream>>>(cnts, BB);

    k_transpose<<<cdiv(512 * 64, 256), 256, 0, stream>>>(P(32), eWihT1, 512, 64);
    k_transpose<<<cdiv(512 * 128, 256), 256, 0, stream>>>(P(33), eWhhT1, 512, 128);
    k_transpose<<<cdiv(512 * 128, 256), 256, 0, stream>>>(P(36), eWihT2, 512, 128);
    k_transpose<<<cdiv(512 * 128, 256), 256, 0, stream>>>(P(37), eWhhT2, 512, 128);
    k_transpose<<<cdiv(512 * 252, 256), 256, 0, stream>>>(P(18), dWihT1, 512, 252);
    k_transpose<<<cdiv(512 * 128, 256), 256, 0, stream>>>(P(19), dWhhT1, 512, 128);
    k_transpose<<<cdiv(512 * 128, 256), 256, 0, stream>>>(P(22), dWihT2, 512, 128);
    k_transpose<<<cdiv(512 * 128, 256), 256, 0, stream>>>(P(23), dWhhT2, 512, 128);
    k_transpose<<<cdiv(240 * 66, 256), 256, 0, stream>>>(P(2), cWihT1, 240, 66);
    k_transpose<<<cdiv(240 * 60, 256), 256, 0, stream>>>(P(3), cWhhT1, 240, 60);
    k_transpose<<<cdiv(240 * 60, 256), 256, 0, stream>>>(P(6), cWihT2, 240, 60);
    k_transpose<<<cdiv(240 * 60, 256), 256, 0, stream>>>(P(7), cWhhT2, 240, 60);
    k_add_vec<<<2, 256, 0, stream>>>(P(34), P(35), bs_e0, 512);
    k_add_vec<<<2, 256, 0, stream>>>(P(38), P(39), bs_e1, 512);
    k_add_vec<<<2, 256, 0, stream>>>(P(20), P(21), bs_d0, 512);
    k_add_vec<<<2, 256, 0, stream>>>(P(24), P(25), bs_d1, 512);
    k_add_vec<<<1, 256, 0, stream>>>(P(4), P(5), bs_c0, 240);
    k_add_vec<<<1, 256, 0, stream>>>(P(8), P(9), bs_c1, 240);

    // ---- enc BN on traffic -> h_bn ----
    bn(stream, stats, traffic, h_bn, P(26), P(27), NN * TT, INF);

    // ---- phase 1: encoder graph-conv per timestep -> hs (segment mean) ----
    zero(stream, hs, (size_t)BB * TT * 128);
    const float* degOut = deg;        // rsqrt(deg_out)
    const float* degIn  = deg + NN;   // rsqrt(deg_in)
    for (int t = 0; t < TT; ++t) {
        zero(stream, mbuf, (size_t)NN * 64);
        scatter(stream, h_bn + (size_t)t * 64, TT * 64, degOut, degIn, src, dst, mbuf, EE, 64);
        gemm(stream, mbuf, 64, P(28), P(29), nullptr, xa, NN, 64, 128, 1);
        zero(stream, mbuf, (size_t)NN * 128);
        scatter(stream, xa, 128, degOut, degIn, src, dst, mbuf, EE, 128);
        gemm(stream, mbuf, 128, P(30), P(31), nullptr, xb, NN, 128, 128, 1);
        k_seg_accum<<<cdiv((long long)NN * 128, 256), 256, 0, stream>>>(
            xb, lbi, cnts, hs, NN, TT, t, 128);
    }

    // ---- phase 2: encoder LSTM over h_bn -> ht (= mean_t of layer2 h) ----
    zero(stream, h1, (size_t)NN * 128); zero(stream, c1, (size_t)NN * 128);
    zero(stream, h2, (size_t)NN * 128); zero(stream, c2, (size_t)NN * 128);
    zero(stream, htacc, (size_t)NN * 128);
    for (int t = 0; t < TT; ++t) {
        gemm(stream, h_bn + (size_t)t * 64, TT * 64, eWihT1, bs_e0, nullptr, gates, NN, 64, 512, 0);
        gemm(stream, h1, 128, eWhhT1, nullptr, gates, gates, NN, 128, 512, 0);
        k_lstm_cell<<<cdiv((long long)NN * 128, 256), 256, 0, stream>>>(
            gates, c1, h1, nullptr, 0, nullptr, NN, 128);
        gemm(stream, h1, 128, eWihT2, bs_e1, nullptr, gates, NN, 128, 512, 0);
        gemm(stream, h2, 128, eWhhT2, nullptr, gates, gates, NN, 128, 512, 0);
        k_lstm_cell<<<cdiv((long long)NN * 128, 256), 256, 0, stream>>>(
            gates, c2, h2, nullptr, 0, htacc, NN, 128);
    }
    k_scale<<<cdiv((long long)NN * 128, 256), 256, 0, stream>>>(
        htacc, ht, 1.0f / (float)TT, (size_t)NN * 128);

    // ---- phase 3: context LSTM (B=64, H=60) -> lt, gtm ----
    k_build_ctx<<<cdiv(BB * TT * 66, 256), 256, 0, stream>>>(
        weeks, minutes, P(51), P(49), ctxin, BB * TT);
    zero(stream, ch1, BB * 60); zero(stream, cc1, BB * 60);
    zero(stream, ch2, BB * 60); zero(stream, cc2, BB * 60);
    for (int t = 0; t < TT; ++t) {
        k_small_gates<<<cdiv(BB * 240, 256), 256, 0, stream>>>(
            ctxin + (size_t)t * 66, TT * 66, 66, cWihT1, ch1, 60, cWhhT1, bs_c0, cg, BB, 240);
        k_lstm_cell<<<cdiv(BB * 60, 256), 256, 0, stream>>>(
            cg, cc1, ch1, nullptr, 0, nullptr, BB, 60);
        k_small_gates<<<cdiv(BB * 240, 256), 256, 0, stream>>>(
            ch1, 60, 60, cWihT2, ch2, 60, cWhhT2, bs_c1, cg, BB, 240);
        k_lstm_cell<<<cdiv(BB * 60, 256), 256, 0, stream>>>(
            cg, cc2, ch2, lt + (size_t)t * 60, TT * 60, nullptr, BB, 60);
    }
    k_mean_T<<<cdiv(BB * 60, 256), 256, 0, stream>>>(lt, gtm, BB, TT, 60);

    // ---- phase 3a: global GCN on region graph ----
    zero(stream, gm, REGF * 64);
    scatter(stream, P(50), 64, gdeg, gdeg + REGF, gsrc, gdst, gm, REF, 64);
    k_small_gemm<<<cdiv(REGF * 64, 256), 256, 0, stream>>>(
        gm, 64, P(44), P(45), gemb, REGF, 64, 64, 1);
    k_gather_rows<<<cdiv(BB * 64, 256), 256, 0, stream>>>(gemb, gsi, gsp, BB, 64);
    k_gather_rows<<<cdiv((long long)NN * 60, 256), 256, 0, stream>>>(gtm, lbi, gth, NN, 60);

    // ---- phase 4: local GCN -> lsh ----
    k_build_xl<<<cdiv((long long)NN * 96, 256), 256, 0, stream>>>(lsf, P(46), lsi, xl, NN);
    zero(stream, mbuf, (size_t)NN * 96);
    scatter(stream, xl, 96, degOut, degIn, src, dst, mbuf, EE, 96);
    gemm(stream, mbuf, 96, P(47), P(48), nullptr, lsh, NN, 96, 64, 1);

    // ---- phase 5: rnn_in = concat(hs, lt, gsp bcast) ; BN dec_bn1 ----
    k_build_rnn_in<<<cdiv(BB * TT * 252, 256), 256, 0, stream>>>(hs, lt, gsp, rnnin, BB, TT);
    bn(stream, stats, rnnin, rnnin, P(10), P(11), BB * TT, 252);

    // ---- phase 6: y = concat(ht, gth, lsh) ; BN dec_bn2 ; dec graph conv ----
    k_build_y<<<cdiv((long long)NN * 252, 256), 256, 0, stream>>>(ht, gth, lsh, ybuf, NN);
    bn(stream, stats, ybuf, ybuf, P(12), P(13), NN, 252);
    zero(stream, mbuf, (size_t)NN * 252);
    scatter(stream, ybuf, 252, degOut, degIn, src, dst, mbuf, EE, 252);
    gemm(stream, mbuf, 252, P(14), P(15), nullptr, xa, NN, 252, 128, 1);
    zero(stream, mbuf, (size_t)NN * 128);
    scatter(stream, xa, 128, degOut, degIn, src, dst, mbuf, EE, 128);
    gemm(stream, mbuf, 128, P(16), P(17), nullptr, xb, NN, 128, 128, 1);
    // xb is now the decoder-conv output ("y" in reference)

    // ---- phase 7: decoder LSTM (B=64) over rnn_in -> aseq ----
    zero(stream, dh1, BB * 128); zero(stream, dc1, BB * 128);
    zero(stream, dh2, BB * 128); zero(stream, dc2, BB * 128);
    for (int t = 0; t < TT; ++t) {
        gemm(stream, rnnin + (size_t)t * 252, TT * 252, dWihT1, bs_d0, nullptr, dg, BB, 252, 512, 0);
        gemm(stream, dh1, 128, dWhhT1, nullptr, dg, dg, BB, 128, 512, 0);
        k_lstm_cell<<<cdiv(BB * 128, 256), 256, 0, stream>>>(
            dg, dc1, dh1, nullptr, 0, nullptr, BB, 128);
        gemm(stream, dh1, 128, dWihT2, bs_d1, nullptr, dg, BB, 128, 512, 0);
        gemm(stream, dh2, 128, dWhhT2, nullptr, dg, dg, BB, 128, 512, 0);
        k_lstm_cell<<<cdiv(BB * 128, 256), 256, 0, stream>>>(
            dg, dc2, dh2, aseq + (size_t)t * 128, TT * 128, nullptr, BB, 128);
    }

    // ---- phase 8: ab = concat(alpha[lbi], beta bcast) into d_out ; BN bn1 ----
    k_build_ab<<<cdiv((long long)NN * TT * 256, 256), 256, 0, stream>>>(
        aseq, hs, xb, ht, lbi, out, NN, TT);
    bn(stream, stats, out, out, P(0), P(1), NN * TT, 256);

    // ---- phase 9: fusion MLP, slabbed, residual into d_out ----
    k_tdm_warm<<<1, 256, 0, stream>>>(P(40), 256, 256);  // TDM prefetch of W1
    k_tdm_warm<<<1, 256, 0, stream>>>(P(42), 256, 256);  // TDM prefetch of W2
    const int NT = NN * TT;
    const int SLAB = 24576;
    for (int s0 = 0; s0 < NT; s0 += SLAB) {
        float* abS = out + (size_t)s0 * 256;
        gemm(stream, abS, 256, P(40), P(41), nullptr, fust, SLAB, 256, 256, 1);
        gemm(stream, fust, 256, P(42), P(43), abS, abS, SLAB, 256, 256, 0);
    }
}